// SparseEmbedding_79147657330929
// MI455X (gfx1250) — compile-verified
//
#include <hip/hip_runtime.h>
#include <stdint.h>

// Problem constants from the reference: B=4, S=256, V=50257, D=768.
#define V_DIM 50257u
#define D_DIM 768u

// Native clang vector types so __builtin_nontemporal_load accepts them and
// emits single b128 NT loads.
typedef unsigned int uintx4 __attribute__((ext_vector_type(4)));

// ---------------------------------------------------------------------------
// Helpers
// ---------------------------------------------------------------------------

// Byte offset of a __shared__ object within the workgroup LDS allocation.
// The async-LDS instructions take their LDS address as a byte offset in a VGPR, so we need the
// raw addrspace(3) offset, not the flat/generic pointer.
__device__ __forceinline__ uint32_t lds_off_u32(const void* p) {
  return (uint32_t)(uintptr_t)(const __attribute__((address_space(3))) void*)p;
}

__device__ __forceinline__ void wait_asynccnt0() {
#if __has_builtin(__builtin_amdgcn_s_wait_asynccnt)
  __builtin_amdgcn_s_wait_asynccnt(0);
#else
  asm volatile("s_wait_asynccnt 0x0" ::: "memory");
#endif
}

// Hit test on raw bits: one-hot elements are bit-exact 0.0f / 1.0f, so a
// bitwise OR-reduce + one compare replaces 4 float compares per vector.
__device__ __forceinline__ void check_vec(uintx4 v, unsigned i,
                                          int* __restrict__ row_idx) {
  if ((v.x | v.y | v.z | v.w) != 0u) {
    unsigned o = v.x ? 0u : v.y ? 1u : v.z ? 2u : 3u;  // cold path
    unsigned e = i * 4u + o;
    unsigned row = e / V_DIM;                          // mul-shift div
    row_idx[row] = (int)(e - row * V_DIM);
  }
}

// ---------------------------------------------------------------------------
// Kernel A: flat non-temporal scan of the one-hot stream.
// Finds the single nonzero column of each (b,s) row and records it in row_idx.
// ~206 MB of b128 NT loads; unrolled x4 so each wave keeps 4 loads in flight
// instead of serializing on s_wait_loadcnt 0 every iteration. No atomics:
// exactly one writer per row by the one-hot invariant.
// ---------------------------------------------------------------------------
__global__ __launch_bounds__(256) void onehot_scan(const uintx4* __restrict__ oh4,
                                                   int* __restrict__ row_idx,
                                                   unsigned nvec,
                                                   unsigned total) {
  const unsigned tid = blockIdx.x * blockDim.x + threadIdx.x;
  const unsigned stride = gridDim.x * blockDim.x;

  unsigned i = tid;
  // Unroll-by-4: issue 4 independent b128 NT loads, then test.
  for (; i + 3u * stride < nvec; i += 4u * stride) {
    uintx4 a = __builtin_nontemporal_load(oh4 + i);
    uintx4 b = __builtin_nontemporal_load(oh4 + i + stride);
    uintx4 c = __builtin_nontemporal_load(oh4 + i + 2u * stride);
    uintx4 d = __builtin_nontemporal_load(oh4 + i + 3u * stride);
    check_vec(a, i, row_idx);
    check_vec(b, i + stride, row_idx);
    check_vec(c, i + 2u * stride, row_idx);
    check_vec(d, i + 3u * stride, row_idx);
  }
  // Remainder vectors.
  for (; i < nvec; i += stride) {
    uintx4 v = __builtin_nontemporal_load(oh4 + i);
    check_vec(v, i, row_idx);
  }
  // Generic scalar tail (empty for this shape since total % 4 == 0).
  for (unsigned e = nvec * 4u + tid; e < total; e += stride) {
    unsigned s = __builtin_nontemporal_load((const unsigned*)oh4 + e);
    if (s != 0u) {
      unsigned row = e / V_DIM;
      row_idx[row] = (int)(e - row * V_DIM);
    }
  }
}

// ---------------------------------------------------------------------------
// Kernel B: gather weight rows via the CDNA5 async global<->LDS engine.
// One 192-thread block per output row: weight[idx,:] -> LDS -> out[row,:].
// Each lane moves 16 B; 192 lanes * 16 B = 768 floats = one D row.
// Each wave stores exactly the LDS bytes it loaded, so a per-wave
// s_wait_asynccnt 0 is the only synchronization required (no barrier).
// ---------------------------------------------------------------------------
__global__ __launch_bounds__(192) void embed_gather(const float* __restrict__ weight,
                                                    const int* __restrict__ row_idx,
                                                    float* __restrict__ out) {
  __shared__ float s_row[D_DIM];            // 3 KB staging tile
  const unsigned row = blockIdx.x;
  const unsigned t = threadIdx.x;           // 0..191, EXEC all-ones
  int id = row_idx[row];
  if (id < 0) id = 0;                        // defensive clamp

  const float* gsrc = weight + (size_t)(unsigned)id * D_DIM + t * 4u;
  float* gdst = out + (size_t)row * D_DIM + t * 4u;
  const uint32_t lds = lds_off_u32(&s_row[t * 4u]);

  // memory -> LDS (async, tracked by ASYNCcnt)
  asm volatile("global_load_async_to_lds_b128 %0, %1, off"
               :
               : "v"(lds), "v"((uint64_t)(uintptr_t)gsrc)
               : "memory");
  wait_asynccnt0();                          // LDS tile resident for this wave

  // LDS -> memory (async)
  asm volatile("global_store_async_from_lds_b128 %0, %1, off"
               :
               : "v"((uint64_t)(uintptr_t)gdst), "v"(lds)
               : "memory");
  wait_asynccnt0();                          // drain before wave exit
}

// ---------------------------------------------------------------------------
// Launch: scan (bandwidth-bound, ~206 MB) then gather (~6 MB).
// ---------------------------------------------------------------------------
extern "C" void kernel_launch(void* const* d_in, const int* in_sizes, int n_in,
                              void* d_out, int out_size, void* d_ws, size_t ws_size,
                              hipStream_t stream) {
  const float* one_hot = (const float*)d_in[0];   // [B,S,V] f32
  const float* weight  = (const float*)d_in[1];   // [V,D]   f32
  float* out = (float*)d_out;                     // [B,S,D] f32
  int* row_idx = (int*)d_ws;                      // B*S ints of scratch

  const unsigned total = (unsigned)in_sizes[0];   // B*S*V
  const unsigned rows  = total / V_DIM;           // B*S (=1024)
  const unsigned nvec  = total / 4u;              // 16B-vector count (exact here)

  onehot_scan<<<dim3(4096), dim3(256), 0, stream>>>(
      (const uintx4*)one_hot, row_idx, nvec, total);
  embed_gather<<<dim3(rows), dim3(192), 0, stream>>>(weight, row_idx, out);
}